// IDCT_31078383353955
// MI455X (gfx1250) — compile-verified
//
#include <hip/hip_runtime.h>
#include <hip/hip_bf16.h>

typedef __attribute__((ext_vector_type(2))) float v2f;
typedef __attribute__((ext_vector_type(4))) float v4f;
typedef __attribute__((ext_vector_type(8))) float v8f;

#define ROWP 36            // padded LDS row stride (floats): 16B-aligned rows, conflict-free gathers
#define WAVES_PER_BLOCK 8  // 256 threads = 8 wave32

// One wave computes Y = H * X * H^T for a sequence of 32x32 fp32 matrices.
//
// WMMA layouts used (f32 16x16x4, wave32), c = lane&15, h = lane>>4:
//   A-frag (16x4):  reg v in {0,1}: A[c][kbase + 2h + v]
//   B-frag (4x16):  reg v in {0,1}: B[kbase + 2h + v][c]
//   C/D   (16x16):  reg r in 0..7 : D[r + 8h][c]
//
// hf[a][q] = { H[16a+c][4q+2h], H[16a+c][4q+2h+1] } serves simultaneously as
//   A-frag of H tile(a, q>>2) chunk (q&3)   (stage 2)
//   B-frag of H^T tile(q>>2, a) chunk (q&3) (stage 1)
__global__ __launch_bounds__(256, 1) void dct32_wmma_kernel(
    const float* __restrict__ X, const float* __restrict__ Hm,
    float* __restrict__ Y, int nmat)
{
  __shared__ float lds[WAVES_PER_BLOCK][32 * ROWP];

  const int lane = threadIdx.x & 31;
  const int wave = (int)(threadIdx.x >> 5);
  const int c    = lane & 15;
  const int h    = lane >> 4;

  // ---- Load shared DCT basis fragments once (hits cache; 4 KB total) ----
  v2f hf[2][8];
#pragma unroll
  for (int a = 0; a < 2; ++a)
#pragma unroll
    for (int q = 0; q < 8; ++q)
      hf[a][q] = *(const v2f*)(Hm + (16 * a + c) * 32 + 4 * q + 2 * h);

  float* sm = lds[wave];
  const int nwaves = (int)gridDim.x * WAVES_PER_BLOCK;
  const int wid    = (int)blockIdx.x * WAVES_PER_BLOCK + wave;

  for (int m = wid; m < nmat; m += nwaves) {
    const float* xm = X + (size_t)m * 1024;
    float*       ym = Y + (size_t)m * 1024;

    // ---- Stage X into LDS: fully coalesced b128 loads/stores ----
#pragma unroll
    for (int t = 0; t < 8; ++t) {
      int f   = t * 32 + lane;      // float4 index 0..255
      int row = f >> 3;             // 8 float4 per 32-float row
      int col = (f & 7) << 2;
      *(v4f*)(sm + row * ROWP + col) = *(const v4f*)(xm + 4 * f);
    }
    // same-wave LDS RAW: all lanes' ds_stores complete when DScnt==0
    asm volatile("s_wait_dscnt 0" ::: "memory");

    // ---- Gather X as A-fragments: xf[i][q] = { X[16i+c][4q+2h], +1 } ----
    v2f xf[2][8];
#pragma unroll
    for (int i = 0; i < 2; ++i)
#pragma unroll
      for (int q = 0; q < 8; ++q)
        xf[i][q] = *(const v2f*)(sm + (16 * i + c) * ROWP + 4 * q + 2 * h);

    // ---- Stage 1: T = X * H^T  (32 x v_wmma_f32_16x16x4_f32) ----
    v8f Tc[2][2];
#pragma unroll
    for (int i = 0; i < 2; ++i)
#pragma unroll
      for (int l = 0; l < 2; ++l) {
        v8f acc = {0.f, 0.f, 0.f, 0.f, 0.f, 0.f, 0.f, 0.f};
#pragma unroll
        for (int p = 0; p < 8; ++p)
          acc = __builtin_amdgcn_wmma_f32_16x16x4_f32(
              false, xf[i][p], false, hf[l][p], (short)0, acc, false, false);
        Tc[i][l] = acc;
      }

    // ---- Stage 2: Y = H * T  (32 WMMAs) ----
    // B-frag of T tile(j,l) chunk kk from C/D regs of Tc[j][l]:
    //   want T[16j + 4kk + 2h + v][16l + c]; column==lane matches C/D, so it
    //   is only a register select + half-wave swap (shfl_xor 16).
    v8f Yc[2][2];
#pragma unroll
    for (int i = 0; i < 2; ++i)
#pragma unroll
      for (int l = 0; l < 2; ++l)
        Yc[i][l] = v8f{0.f, 0.f, 0.f, 0.f, 0.f, 0.f, 0.f, 0.f};

#pragma unroll
    for (int l = 0; l < 2; ++l)
#pragma unroll
      for (int j = 0; j < 2; ++j)
#pragma unroll
        for (int kk = 0; kk < 4; ++kk) {
          v2f tb;
#pragma unroll
          for (int v = 0; v < 2; ++v) {
            float a = Tc[j][l][(4 * kk + v) & 7];       // row 4kk+v     (mod 8)
            float b = Tc[j][l][(4 * kk + 2 + v) & 7];   // row 4kk+2+v   (mod 8)
            float r;
            if (kk < 2) r = (h == 0) ? a : __shfl_xor(b, 16, 32);
            else        r = (h == 0) ? __shfl_xor(a, 16, 32) : b;
            tb[v] = r;
          }
          const int p = 4 * j + kk;
          Yc[0][l] = __builtin_amdgcn_wmma_f32_16x16x4_f32(
              false, hf[0][p], false, tb, (short)0, Yc[0][l], false, false);
          Yc[1][l] = __builtin_amdgcn_wmma_f32_16x16x4_f32(
              false, hf[1][p], false, tb, (short)0, Yc[1][l], false, false);
        }

    // ---- Store Y (C/D layout -> row-major): 64B contiguous per half-wave ----
#pragma unroll
    for (int i = 0; i < 2; ++i)
#pragma unroll
      for (int l = 0; l < 2; ++l)
#pragma unroll
        for (int r = 0; r < 8; ++r)
          ym[(16 * i + r + 8 * h) * 32 + 16 * l + c] = Yc[i][l][r];
  }
}

extern "C" void kernel_launch(void* const* d_in, const int* in_sizes, int n_in,
                              void* d_out, int out_size, void* d_ws, size_t ws_size,
                              hipStream_t stream) {
  const float* x = (const float*)d_in[0];   // (B, C, 32, 32) fp32
  const float* H = (const float*)d_in[1];   // (32, 32) fp32
  float* out = (float*)d_out;

  const int nmat = in_sizes[0] / 1024;      // B*C matrices of 32x32
  int blocks = (nmat + WAVES_PER_BLOCK - 1) / WAVES_PER_BLOCK;
  if (blocks > 2048) blocks = 2048;         // grid-stride: ~4 matrices/wave at 65536
  if (blocks < 1) blocks = 1;

  dct32_wmma_kernel<<<blocks, 256, 0, stream>>>(x, H, out, nmat);
}